// GATNet_63548336112326
// MI455X (gfx1250) — compile-verified
//
#include <hip/hip_runtime.h>
#include <hip/hip_bf16.h>

typedef float v2f __attribute__((ext_vector_type(2)));
typedef float v8f __attribute__((ext_vector_type(8)));

#define NEG_SLOPE 0.2f

// ---------------------------------------------------------------------------
// utility: fill
// ---------------------------------------------------------------------------
__global__ void fill_kernel(float* __restrict__ p, int n, float v) {
    int i = blockIdx.x * blockDim.x + threadIdx.x;
    if (i < n) p[i] = v;
}

// monotone float atomic-max via int max (v>=0) / uint min (v<0); init = -inf
__device__ inline void atomicMaxF(float* addr, float v) {
    if (v >= 0.0f) atomicMax((int*)addr, __float_as_int(v));
    else           atomicMin((unsigned int*)addr, __float_as_uint(v));
}

// ---------------------------------------------------------------------------
// GEMM1: h1[N,128] = x[N,128] @ W1[128,128]  via V_WMMA_F32_16X16X4_F32
// one wave per 16x16 output tile; block = 8 waves = 8 column tiles (full 128)
// ---------------------------------------------------------------------------
__global__ __launch_bounds__(256) void gemm1_wmma_kernel(
    const float* __restrict__ x, const float* __restrict__ W1,
    float* __restrict__ h1, int n_nodes) {
    const int lane = threadIdx.x & 31;
    const int wave = threadIdx.x >> 5;       // n-tile 0..7
    const int m16  = blockIdx.x;             // m-tile
    const int l15  = lane & 15;
    const int kOff = (lane >> 4) * 2;        // lanes 16-31 hold K+2,K+3

    int row = m16 * 16 + l15;
    if (row > n_nodes - 1) row = n_nodes - 1;      // never triggers: N%16==0
    const int col = wave * 16 + l15;

    v8f acc = {};
    #pragma unroll 4
    for (int k0 = 0; k0 < 128; k0 += 4) {
        v2f a, b;
        // A 16x4 (MxK): VGPR0=K(kOff), VGPR1=K(kOff+1) -> contiguous pair
        a.x = x[row * 128 + k0 + kOff + 0];
        a.y = x[row * 128 + k0 + kOff + 1];
        // B 4x16 (KxN): row K striped across lanes within a VGPR
        b.x = W1[(k0 + kOff + 0) * 128 + col];
        b.y = W1[(k0 + kOff + 1) * 128 + col];
        acc = __builtin_amdgcn_wmma_f32_16x16x4_f32(
            /*neg_a=*/false, a, /*neg_b=*/false, b,
            /*c_mod=*/(short)0, acc, /*reuse_a=*/false, /*reuse_b=*/false);
    }
    // D 16x16 f32: VGPR r -> M=r (lanes 0-15) / M=r+8 (lanes 16-31), N=lane&15
    const int rowBase = m16 * 16 + ((lane >> 4) ? 8 : 0);
    #pragma unroll
    for (int r = 0; r < 8; ++r) {
        const int orow = rowBase + r;
        if (orow < n_nodes) h1[orow * 128 + col] = acc[r];
    }
}

// ---------------------------------------------------------------------------
// attention terms layer 1: a_s1[n,h] = sum_c h1[n,h*32+c]*a_src1[h,c]
// one wave per node; lane covers 4 channels; 8 lanes per head
// ---------------------------------------------------------------------------
__global__ __launch_bounds__(256) void attn1_kernel(
    const float* __restrict__ h1, const float* __restrict__ a_src,
    const float* __restrict__ a_dst, float* __restrict__ as_out,
    float* __restrict__ ad_out, int n_nodes) {
    const int node = (int)((blockIdx.x * (unsigned)blockDim.x + threadIdx.x) >> 5);
    const int lane = threadIdx.x & 31;
    if (node >= n_nodes) return;
    const int c4 = lane * 4;
    const float4 hv = *(const float4*)&h1[(size_t)node * 128 + c4];
    const float4 sw = *(const float4*)&a_src[c4];
    const float4 dw = *(const float4*)&a_dst[c4];
    float ps = hv.x * sw.x + hv.y * sw.y + hv.z * sw.z + hv.w * sw.w;
    float pd = hv.x * dw.x + hv.y * dw.y + hv.z * dw.z + hv.w * dw.w;
    #pragma unroll
    for (int off = 1; off < 8; off <<= 1) {
        ps += __shfl_xor(ps, off, 32);
        pd += __shfl_xor(pd, off, 32);
    }
    if ((lane & 7) == 0) {
        const int hd = lane >> 3;
        as_out[node * 4 + hd] = ps;
        ad_out[node * 4 + hd] = pd;
    }
}

// ---------------------------------------------------------------------------
// edge segment-max  (generic over H heads); self-loops appended at [E, E+N)
// ---------------------------------------------------------------------------
__global__ void edge_max_kernel(const int* __restrict__ ei, int nE, int n_nodes, int H,
                                const float* __restrict__ a_s,
                                const float* __restrict__ a_d,
                                float* __restrict__ m) {
    const int t = blockIdx.x * blockDim.x + threadIdx.x;
    const int total = (nE + n_nodes) * H;
    if (t >= total) return;
    const int e  = t / H;
    const int hd = t - e * H;
    const int src = (e < nE) ? ei[e]      : (e - nE);
    const int dst = (e < nE) ? ei[nE + e] : (e - nE);
    float v = a_s[src * H + hd] + a_d[dst * H + hd];
    v = (v >= 0.0f) ? v : NEG_SLOPE * v;
    atomicMaxF(&m[dst * H + hd], v);
}

// edge exp-sum (softmax denominator)
__global__ void edge_expsum_kernel(const int* __restrict__ ei, int nE, int n_nodes, int H,
                                   const float* __restrict__ a_s,
                                   const float* __restrict__ a_d,
                                   const float* __restrict__ m,
                                   float* __restrict__ denom) {
    const int t = blockIdx.x * blockDim.x + threadIdx.x;
    const int total = (nE + n_nodes) * H;
    if (t >= total) return;
    const int e  = t / H;
    const int hd = t - e * H;
    const int src = (e < nE) ? ei[e]      : (e - nE);
    const int dst = (e < nE) ? ei[nE + e] : (e - nE);
    float v = a_s[src * H + hd] + a_d[dst * H + hd];
    v = (v >= 0.0f) ? v : NEG_SLOPE * v;
    atomicAdd(&denom[dst * H + hd], expf(v - m[dst * H + hd]));
}

// ---------------------------------------------------------------------------
// layer-1 aggregation: wave per edge; lane covers 4 channels (head = lane/8)
// agg1[dst] += h1[src] * alpha[head]
// ---------------------------------------------------------------------------
__global__ __launch_bounds__(256) void edge_agg1_kernel(
    const int* __restrict__ ei, int nE, int n_nodes,
    const float* __restrict__ a_s, const float* __restrict__ a_d,
    const float* __restrict__ m, const float* __restrict__ denom,
    const float* __restrict__ h1, float* __restrict__ agg) {
    const int edge = (int)((blockIdx.x * (unsigned)blockDim.x + threadIdx.x) >> 5);
    const int lane = threadIdx.x & 31;
    const int total = nE + n_nodes;
    if (edge >= total) return;
    const int src = (edge < nE) ? ei[edge]      : (edge - nE);
    const int dst = (edge < nE) ? ei[nE + edge] : (edge - nE);
    const int hd  = lane >> 3;
    float v = a_s[src * 4 + hd] + a_d[dst * 4 + hd];
    v = (v >= 0.0f) ? v : NEG_SLOPE * v;
    const float alpha = expf(v - m[dst * 4 + hd]) / (denom[dst * 4 + hd] + 1e-16f);
    const int c4 = lane * 4;
    const float4 hv = *(const float4*)&h1[(size_t)src * 128 + c4];
    float* o = &agg[(size_t)dst * 128 + c4];
    atomicAdd(o + 0, hv.x * alpha);
    atomicAdd(o + 1, hv.y * alpha);
    atomicAdd(o + 2, hv.z * alpha);
    atomicAdd(o + 3, hv.w * alpha);
}

// ---------------------------------------------------------------------------
// fused: out1 = relu(agg1 + b1); o2 = out1 @ W2[128,2]; attn scalars layer 2
// one wave per node, full-wave shfl reduction
// ---------------------------------------------------------------------------
__global__ __launch_bounds__(256) void node2_kernel(
    const float* __restrict__ agg1, const float* __restrict__ b1,
    const float* __restrict__ W2, const float* __restrict__ a_src2,
    const float* __restrict__ a_dst2, float* __restrict__ o2,
    float* __restrict__ as2, float* __restrict__ ad2, int n_nodes) {
    const int node = (int)((blockIdx.x * (unsigned)blockDim.x + threadIdx.x) >> 5);
    const int lane = threadIdx.x & 31;
    if (node >= n_nodes) return;
    const int c4 = lane * 4;
    const float4 g  = *(const float4*)&agg1[(size_t)node * 128 + c4];
    const float4 bb = *(const float4*)&b1[c4];
    const float r0 = fmaxf(g.x + bb.x, 0.0f);
    const float r1 = fmaxf(g.y + bb.y, 0.0f);
    const float r2 = fmaxf(g.z + bb.z, 0.0f);
    const float r3 = fmaxf(g.w + bb.w, 0.0f);
    float p0 = r0 * W2[(c4 + 0) * 2 + 0] + r1 * W2[(c4 + 1) * 2 + 0]
             + r2 * W2[(c4 + 2) * 2 + 0] + r3 * W2[(c4 + 3) * 2 + 0];
    float p1 = r0 * W2[(c4 + 0) * 2 + 1] + r1 * W2[(c4 + 1) * 2 + 1]
             + r2 * W2[(c4 + 2) * 2 + 1] + r3 * W2[(c4 + 3) * 2 + 1];
    #pragma unroll
    for (int off = 1; off < 32; off <<= 1) {
        p0 += __shfl_xor(p0, off, 32);
        p1 += __shfl_xor(p1, off, 32);
    }
    if (lane == 0) {
        o2[node * 2 + 0] = p0;
        o2[node * 2 + 1] = p1;
        as2[node] = p0 * a_src2[0] + p1 * a_src2[1];
        ad2[node] = p0 * a_dst2[0] + p1 * a_dst2[1];
    }
}

// layer-2 aggregation: thread per edge (H=1, C=2)
__global__ void edge_agg2_kernel(const int* __restrict__ ei, int nE, int n_nodes,
                                 const float* __restrict__ as2,
                                 const float* __restrict__ ad2,
                                 const float* __restrict__ m2,
                                 const float* __restrict__ den2,
                                 const float* __restrict__ o2,
                                 float* __restrict__ agg2) {
    const int e = blockIdx.x * blockDim.x + threadIdx.x;
    if (e >= nE + n_nodes) return;
    const int src = (e < nE) ? ei[e]      : (e - nE);
    const int dst = (e < nE) ? ei[nE + e] : (e - nE);
    float v = as2[src] + ad2[dst];
    v = (v >= 0.0f) ? v : NEG_SLOPE * v;
    const float alpha = expf(v - m2[dst]) / (den2[dst] + 1e-16f);
    atomicAdd(&agg2[dst * 2 + 0], o2[src * 2 + 0] * alpha);
    atomicAdd(&agg2[dst * 2 + 1], o2[src * 2 + 1] * alpha);
}

// bias + log_softmax over 2 channels
__global__ void final_kernel(const float* __restrict__ agg2,
                             const float* __restrict__ b2,
                             float* __restrict__ out, int n_nodes) {
    const int n = blockIdx.x * blockDim.x + threadIdx.x;
    if (n >= n_nodes) return;
    const float v0 = agg2[n * 2 + 0] + b2[0];
    const float v1 = agg2[n * 2 + 1] + b2[1];
    const float mx  = fmaxf(v0, v1);
    const float lse = mx + logf(expf(v0 - mx) + expf(v1 - mx));
    out[n * 2 + 0] = v0 - lse;
    out[n * 2 + 1] = v1 - lse;
}

// ---------------------------------------------------------------------------
extern "C" void kernel_launch(void* const* d_in, const int* in_sizes, int n_in,
                              void* d_out, int out_size, void* d_ws, size_t ws_size,
                              hipStream_t stream) {
    const float* x      = (const float*)d_in[0];
    const int*   ei     = (const int*)  d_in[1];
    const float* W1     = (const float*)d_in[2];
    const float* a_src1 = (const float*)d_in[3];
    const float* a_dst1 = (const float*)d_in[4];
    const float* b1     = (const float*)d_in[5];
    const float* W2     = (const float*)d_in[6];
    const float* a_src2 = (const float*)d_in[7];
    const float* a_dst2 = (const float*)d_in[8];
    const float* b2     = (const float*)d_in[9];
    float* out = (float*)d_out;

    const int N  = in_sizes[0] / 128;     // 50000
    const int nE = in_sizes[1] / 2;       // 800000
    const int T  = nE + N;                // edges incl. self-loops

    // workspace layout (floats)
    float* ws   = (float*)d_ws;
    float* h1   = ws;                             // N*128
    float* agg1 = h1   + (size_t)N * 128;         // N*128
    float* as1  = agg1 + (size_t)N * 128;         // N*4
    float* ad1  = as1  + (size_t)N * 4;           // N*4
    float* m1   = ad1  + (size_t)N * 4;           // N*4
    float* den1 = m1   + (size_t)N * 4;           // N*4
    float* o2   = den1 + (size_t)N * 4;           // N*2
    float* as2  = o2   + (size_t)N * 2;           // N
    float* ad2  = as2  + (size_t)N;               // N
    float* m2   = ad2  + (size_t)N;               // N
    float* den2 = m2   + (size_t)N;               // N
    float* agg2 = den2 + (size_t)N;               // N*2

    const float NEGINF = -__builtin_inff();
    auto cdiv = [](int a, int b) { return (a + b - 1) / b; };

    // init accumulators every call (harness does not re-poison)
    fill_kernel<<<cdiv(N * 128, 256), 256, 0, stream>>>(agg1, N * 128, 0.0f);
    fill_kernel<<<cdiv(N * 4, 256),   256, 0, stream>>>(m1,   N * 4, NEGINF);
    fill_kernel<<<cdiv(N * 4, 256),   256, 0, stream>>>(den1, N * 4, 0.0f);
    fill_kernel<<<cdiv(N, 256),       256, 0, stream>>>(m2,   N, NEGINF);
    fill_kernel<<<cdiv(N, 256),       256, 0, stream>>>(den2, N, 0.0f);
    fill_kernel<<<cdiv(N * 2, 256),   256, 0, stream>>>(agg2, N * 2, 0.0f);

    // layer 1
    gemm1_wmma_kernel<<<cdiv(N, 16), 256, 0, stream>>>(x, W1, h1, N);
    attn1_kernel<<<cdiv(N, 8), 256, 0, stream>>>(h1, a_src1, a_dst1, as1, ad1, N);
    edge_max_kernel   <<<cdiv(T * 4, 256), 256, 0, stream>>>(ei, nE, N, 4, as1, ad1, m1);
    edge_expsum_kernel<<<cdiv(T * 4, 256), 256, 0, stream>>>(ei, nE, N, 4, as1, ad1, m1, den1);
    edge_agg1_kernel  <<<cdiv(T, 8),       256, 0, stream>>>(ei, nE, N, as1, ad1, m1, den1, h1, agg1);

    // fused relu + GEMM2 + layer-2 attention scalars
    node2_kernel<<<cdiv(N, 8), 256, 0, stream>>>(agg1, b1, W2, a_src2, a_dst2, o2, as2, ad2, N);

    // layer 2
    edge_max_kernel   <<<cdiv(T, 256), 256, 0, stream>>>(ei, nE, N, 1, as2, ad2, m2);
    edge_expsum_kernel<<<cdiv(T, 256), 256, 0, stream>>>(ei, nE, N, 1, as2, ad2, m2, den2);
    edge_agg2_kernel  <<<cdiv(T, 256), 256, 0, stream>>>(ei, nE, N, as2, ad2, m2, den2, o2, agg2);

    final_kernel<<<cdiv(N, 256), 256, 0, stream>>>(agg2, b2, out, N);
}